// test_CustomLayer_77378130804966
// MI455X (gfx1250) — compile-verified
//
#include <hip/hip_runtime.h>

// ---------------------------------------------------------------------------
// RGB -> CIELAB + per-image L min/max normalization, fused single pass.
// One workgroup per image. The 192KB image streams into CDNA5's 320KB LDS via
// FOUR pipelined Tensor Data Mover descriptors (48KB each); compute on chunk i
// overlaps the DMA of chunks i+1..3 via graded s_wait_tensorcnt. Lab overwrites
// sRGB in place (b128 LDS traffic), L min/max reduces on-chip (wave32 shuffles),
// and the normalized result stores out as b128. HBM traffic = 1x read + 1x
// write = ~201MB -> ~8.6us roofline @ 23.3TB/s, with DMA/compute overlapped.
// ---------------------------------------------------------------------------

typedef __attribute__((ext_vector_type(4))) unsigned int u32x4;
typedef __attribute__((ext_vector_type(8))) int          i32x8;
typedef __attribute__((ext_vector_type(4))) int          i32x4;

#define PIXELS          16384      // 128*128
#define FLOATS_PER_IMG  49152      // PIXELS*3
#define BYTES_PER_IMG   196608     // FLOATS_PER_IMG*4
#define THREADS         512
#define NCHUNK          4
#define CHUNK_FLOATS    12288      // FLOATS_PER_IMG / NCHUNK
#define CHUNK_BYTES     49152
#define QUADS_PER_CHUNK 1024       // (PIXELS/4) / NCHUNK ... 4096 pixels/chunk
#define QUADS_TOTAL     4096       // PIXELS / 4

// raw hardware transcendentals: v_exp_f32 / v_log_f32 (both base-2)
__device__ __forceinline__ float hw_exp2(float x) { return __builtin_amdgcn_exp2f(x); }
__device__ __forceinline__ float hw_log2(float x) { return __builtin_amdgcn_logf(x); }

// inverse sRGB gamma: c > 0.04045 ? ((c+0.055)/1.055)^2.4 : c/12.92
__device__ __forceinline__ float srgb_inv(float c) {
    float p = hw_exp2(2.4f * hw_log2((c + 0.055f) * (1.0f / 1.055f)));
    return c > 0.04045f ? p : c * (1.0f / 12.92f);
}

// CIE Lab nonlinearity: t > eps ? cbrt(t) : 7.787t + 16/116
__device__ __forceinline__ float lab_f(float t) {
    float cb = hw_exp2((1.0f / 3.0f) * hw_log2(t));
    return t > 0.008856f ? cb : 7.787f * t + (16.0f / 116.0f);
}

// sRGB (r,g,b) -> Lab (L,a,b) in place; D65 white folded into the matrix rows
__device__ __forceinline__ void lab_pixel(float& r, float& g, float& b) {
    float lr = srgb_inv(r), lg = srgb_inv(g), lb = srgb_inv(b);
    float fx = lab_f(lr * (0.412453f / 0.95047f) + lg * (0.357580f / 0.95047f) + lb * (0.180423f / 0.95047f));
    float fy = lab_f(lr * 0.212671f              + lg * 0.715160f              + lb * 0.072169f);
    float fz = lab_f(lr * (0.019334f / 1.08883f) + lg * (0.119193f / 1.08883f) + lb * (0.950227f / 1.08883f));
    r = 116.0f * fy - 16.0f;
    g = 500.0f * (fx - fy);
    b = 200.0f * (fy - fz);
}

#if __has_builtin(__builtin_amdgcn_tensor_load_to_lds)
// One 48KB 1-D TDM descriptor: global image chunk -> LDS chunk.  -> TENSORcnt
__device__ __forceinline__ void tdm_load_chunk(const float* in, int img, int chunk) {
    unsigned long long ga = (unsigned long long)(const void*)in
                          + (unsigned long long)img * BYTES_PER_IMG
                          + (unsigned long long)chunk * CHUNK_BYTES;
    // D# group0: count=1, lds_addr=chunk*48KB, 57-bit global addr, type=2
    u32x4 g0 = { 1u,
                 (unsigned)(chunk * CHUNK_BYTES),
                 (unsigned)(ga & 0xFFFFFFFFull),
                 (unsigned)((ga >> 32) & 0x1FFFFFFull) | (2u << 30) };
    // D# group1: no multicast, data_size=4B, 1-D tensor/tile of 12288 elems
    i32x8 g1 = { (int)(2u << 16),                               // data_size=2 (4B)
                 (int)((CHUNK_FLOATS & 0xFFFFu) << 16),         // tensor_dim0 lo16
                 (int)(((unsigned)CHUNK_FLOATS >> 16) | (1u << 16)), // dim0 hi, tensor_dim1=1
                 (int)((CHUNK_FLOATS & 0xFFFFu) << 16),         // tile_dim0 = 12288
                 0,                                             // tile_dim1/2 unused
                 (int)CHUNK_FLOATS,                             // tensor_dim0_stride lo
                 0, 0 };
    i32x4 g2 = { 0, 0, 0, 0 };                                  // <=2D: groups 2/3 null
    i32x4 g3 = { 0, 0, 0, 0 };
    i32x8 gpad = { 0, 0, 0, 0, 0, 0, 0, 0 };
    __builtin_amdgcn_tensor_load_to_lds(g0, g1, g2, g3, gpad, 0);
}
#endif

// Convert one 48KB chunk in LDS from sRGB to Lab in place (b128 traffic),
// accumulating this thread's L min/max. 2 pixel-quads (24 floats) per thread.
__device__ __forceinline__ void compute_chunk(float* smem, int chunk, int tid,
                                              float& mn, float& mx) {
    float4* sm4 = (float4*)smem;
#pragma unroll
    for (int j = 0; j < 2; ++j) {
        const int q = chunk * QUADS_PER_CHUNK + j * THREADS + tid;   // quad index
        float4 v0 = sm4[q * 3 + 0];
        float4 v1 = sm4[q * 3 + 1];
        float4 v2 = sm4[q * 3 + 2];
        float p[12] = { v0.x, v0.y, v0.z, v0.w, v1.x, v1.y, v1.z, v1.w,
                        v2.x, v2.y, v2.z, v2.w };
#pragma unroll
        for (int t = 0; t < 4; ++t) {
            lab_pixel(p[3 * t], p[3 * t + 1], p[3 * t + 2]);
            mn = fminf(mn, p[3 * t]);
            mx = fmaxf(mx, p[3 * t]);
        }
        sm4[q * 3 + 0] = make_float4(p[0], p[1], p[2],  p[3]);
        sm4[q * 3 + 1] = make_float4(p[4], p[5], p[6],  p[7]);
        sm4[q * 3 + 2] = make_float4(p[8], p[9], p[10], p[11]);
    }
}

__global__ __launch_bounds__(THREADS)
void rgb2lab_norm_kernel(const float* __restrict__ in, float* __restrict__ out) {
    extern __shared__ float smem[];        // [0,49152): image; [49152,+32): scratch
    const int tid = threadIdx.x;
    const int img = blockIdx.x;

    float mn = 3.402823466e38f, mx = -3.402823466e38f;

    // ---- Pipelined TDM fill + in-place sRGB->Lab ---------------------------
#if __has_builtin(__builtin_amdgcn_tensor_load_to_lds)
    if (tid < 32) {   // one wave owns the DMA descriptors (in-order completion)
        tdm_load_chunk(in, img, 0);
        tdm_load_chunk(in, img, 1);
        tdm_load_chunk(in, img, 2);
        tdm_load_chunk(in, img, 3);
    }
    if (tid < 32) __builtin_amdgcn_s_wait_tensorcnt(3);  // chunk 0 landed
    __syncthreads();
    compute_chunk(smem, 0, tid, mn, mx);                 // overlaps DMA of 1..3
    if (tid < 32) __builtin_amdgcn_s_wait_tensorcnt(2);  // chunk 1 landed
    __syncthreads();
    compute_chunk(smem, 1, tid, mn, mx);
    if (tid < 32) __builtin_amdgcn_s_wait_tensorcnt(1);  // chunk 2 landed
    __syncthreads();
    compute_chunk(smem, 2, tid, mn, mx);
    if (tid < 32) __builtin_amdgcn_s_wait_tensorcnt(0);  // chunk 3 landed
    __syncthreads();
    compute_chunk(smem, 3, tid, mn, mx);
#else
    {   // fallback: cooperative vectorized copy then compute
        const float4* src = (const float4*)(in + (unsigned long long)img * FLOATS_PER_IMG);
        float4* dst = (float4*)smem;
        for (int i = tid; i < FLOATS_PER_IMG / 4; i += THREADS) dst[i] = src[i];
        __syncthreads();
        for (int c = 0; c < NCHUNK; ++c) compute_chunk(smem, c, tid, mn, mx);
    }
#endif

    // ---- Block min/max of L (wave32 shuffle + LDS scratch) -----------------
#pragma unroll
    for (int off = 16; off > 0; off >>= 1) {
        mn = fminf(mn, __shfl_xor(mn, off, 32));
        mx = fmaxf(mx, __shfl_xor(mx, off, 32));
    }
    float* red = smem + FLOATS_PER_IMG;          // 32-float scratch
    if ((tid & 31) == 0) { red[tid >> 5] = mn; red[16 + (tid >> 5)] = mx; }
    __syncthreads();
    float Lmin = red[0], Lmax = red[16];
#pragma unroll
    for (int w = 1; w < 16; ++w) {
        Lmin = fminf(Lmin, red[w]);
        Lmax = fmaxf(Lmax, red[16 + w]);
    }
    const float inv = 1.0f / (Lmax - Lmin);
    const float sAB = 1.0f / 255.0f;

    // ---- Normalize & store: 3x b128 per quad, 8 quads per thread -----------
    float4* sm4 = (float4*)smem;
    float4* o4  = (float4*)(out + (unsigned long long)img * FLOATS_PER_IMG);
#pragma unroll
    for (int k = 0; k < QUADS_TOTAL / THREADS; ++k) {
        const int q = k * THREADS + tid;
        float4 v0 = sm4[q * 3 + 0];
        float4 v1 = sm4[q * 3 + 1];
        float4 v2 = sm4[q * 3 + 2];
        // 12 floats = 4 pixels [L,a,b]; L at positions 0,3,6,9
        v0.x = (v0.x - Lmin) * inv;  v0.y = (v0.y + 128.0f) * sAB;
        v0.z = (v0.z + 128.0f) * sAB; v0.w = (v0.w - Lmin) * inv;
        v1.x = (v1.x + 128.0f) * sAB; v1.y = (v1.y + 128.0f) * sAB;
        v1.z = (v1.z - Lmin) * inv;  v1.w = (v1.w + 128.0f) * sAB;
        v2.x = (v2.x + 128.0f) * sAB; v2.y = (v2.y - Lmin) * inv;
        v2.z = (v2.z + 128.0f) * sAB; v2.w = (v2.w + 128.0f) * sAB;
        o4[q * 3 + 0] = v0;
        o4[q * 3 + 1] = v1;
        o4[q * 3 + 2] = v2;
    }
}

extern "C" void kernel_launch(void* const* d_in, const int* in_sizes, int n_in,
                              void* d_out, int out_size, void* d_ws, size_t ws_size,
                              hipStream_t stream) {
    (void)n_in; (void)d_ws; (void)ws_size; (void)out_size;
    const float* in = (const float*)d_in[0];
    float* out = (float*)d_out;
    const int images = in_sizes[0] / FLOATS_PER_IMG;            // 512
    const size_t shmem = (FLOATS_PER_IMG + 32) * sizeof(float); // 196736 B < 320KB
    rgb2lab_norm_kernel<<<images, THREADS, shmem, stream>>>(in, out);
}